// EncoderLayer_28595892256972
// MI455X (gfx1250) — compile-verified
//
#include <hip/hip_runtime.h>

typedef __attribute__((ext_vector_type(2))) float v2f;
typedef __attribute__((ext_vector_type(8))) float v8f;

#define B_      16
#define N_IN    16384
#define N_OUT   8192
#define IDIM    128
#define ODIM    128
#define KDIM    256        // 2*IDIM
#define NDIR    6
#define TILES_MAX (N_OUT / 16)   // 512 worst-case 16-row tiles per bucket

// ---------------- kernel 0: zero bucket counters (must run every call) ----------
__global__ void k_zero_counts(int* counts) {
    int t = threadIdx.x;
    if (t < B_ * NDIR) counts[t] = 0;
}

// ---------------- kernel 1: transpose W -> Wt[d][col][k] ------------------------
// Makes the WMMA B-fragment (two consecutive K values for a fixed col) one b64 load.
__global__ void k_transpose_w(const float* __restrict__ W, float* __restrict__ Wt) {
    int idx = blockIdx.x * 256 + threadIdx.x;            // NDIR*KDIM*ODIM = 196608
    if (idx >= NDIR * KDIM * ODIM) return;
    int d   = idx / (KDIM * ODIM);
    int rem = idx % (KDIM * ODIM);
    int k   = rem / ODIM;
    int c   = rem % ODIM;
    Wt[((size_t)d * ODIM + c) * KDIM + k] = W[idx];
}

// ---------------- kernel 2: bucket rows by (batch, direction) -------------------
__global__ void k_bucketize(const int* __restrict__ vec, const int* __restrict__ dmap,
                            int* __restrict__ counts, int* __restrict__ buckets) {
    int idx = blockIdx.x * 256 + threadIdx.x;            // B_*N_OUT = 131072
    if (idx >= B_ * N_OUT) return;
    int b = idx / N_OUT;
    int n = idx % N_OUT;
    int d = dmap[vec[idx]];
    int pos = atomicAdd(&counts[b * NDIR + d], 1);
    buckets[(size_t)(b * NDIR + d) * N_OUT + pos] = n;
}

// ---------------- kernel 3: gather + WMMA f32 grouped GEMM ----------------------
// One block = 16 rows of one (b,d) bucket, full ODIM=128 columns.
// 8 wave32s; wave w computes the 16x16 tile for cols [16w, 16w+16).
__global__ __launch_bounds__(256, 1)
void k_gemm_tile(const float* __restrict__ last, const float* __restrict__ Wt,
                 const float* __restrict__ bias, const float* __restrict__ alpha_p,
                 const int* __restrict__ vec, const int* __restrict__ drev,
                 const int* __restrict__ child_l, const int* __restrict__ child_r,
                 const int* __restrict__ counts, const int* __restrict__ buckets,
                 float* __restrict__ out) {
    __shared__ float pairS[16 * KDIM];   // 16 KB: 16 gathered pair rows
    __shared__ int   nrowS[16];

    const int bd   = blockIdx.x / TILES_MAX;     // b*NDIR + d
    const int tile = blockIdx.x % TILES_MAX;
    const int cnt  = counts[bd];
    const int row0 = tile * 16;
    if (row0 >= cnt) return;                     // empty tile: 1 SGPR load + exit
    const int b = bd / NDIR;
    const int d = bd % NDIR;
    const int nrows = min(16, cnt - row0);

    const int tid = threadIdx.x;

    if (tid < 16) {
        // pad partial tiles by replicating a valid row (stores are masked later)
        int r = (tid < nrows) ? tid : 0;
        nrowS[tid] = buckets[(size_t)bd * N_OUT + row0 + r];
    }
    __syncthreads();

    // ---- gather: thread tid -> row tid/16, 16-float segment (tid%16)*16 ----
    {
        int r   = tid >> 4;          // 0..15
        int seg = tid & 15;          // 0..15
        int n   = nrowS[r];
        int rev = drev[vec[b * N_OUT + n]];
        int cl  = child_l[n], cr = child_r[n];
        int srcA = rev ? cr : cl;    // pair[0..127]
        int srcB = rev ? cl : cr;    // pair[128..255]
        int off  = seg * 16;
        const float* src = (off < IDIM)
            ? (last + ((size_t)b * N_IN + srcA) * IDIM + off)
            : (last + ((size_t)b * N_IN + srcB) * IDIM + (off - IDIM));
        float* dst = &pairS[r * KDIM + off];
#pragma unroll
        for (int i = 0; i < 4; ++i)
            ((float4*)dst)[i] = ((const float4*)src)[i];   // global_load_b128 -> ds_store_b128
    }
    __syncthreads();

    // ---- WMMA: K=256 in steps of 4 -> 64 x v_wmma_f32_16x16x4_f32 per wave ----
    const int lane = tid & 31;
    const int wave = tid >> 5;
    const int col0 = wave * 16;
    const int colN = lane & 15;                 // N within tile (A row M likewise)
    const int koff = (lane >> 4) * 2;           // lanes 16-31 hold K+2, K+3
    const int arow = lane & 15;
    const float* wcol = Wt + ((size_t)d * ODIM + (col0 + colN)) * KDIM;

    v8f acc = {};
#pragma unroll 8
    for (int k = 0; k < KDIM; k += 4) {
        v2f a, bf;
        a.x  = pairS[arow * KDIM + k + koff];
        a.y  = pairS[arow * KDIM + k + koff + 1];
        bf.x = wcol[k + koff];
        bf.y = wcol[k + koff + 1];
        acc = __builtin_amdgcn_wmma_f32_16x16x4_f32(
                  false, a, false, bf, (short)0, acc, false, false);
    }

    // ---- epilogue: bias + PReLU, masked scatter per C/D layout ----
    const float alpha = *alpha_p;
    const float bb    = bias[d * ODIM + col0 + colN];
    const int mbase   = (lane >> 4) * 8;        // lanes 16-31 hold M=8..15
#pragma unroll
    for (int r = 0; r < 8; ++r) {
        int row = mbase + r;
        if (row < nrows) {
            float y = acc[r] + bb;
            y = (y >= 0.0f) ? y : alpha * y;
            out[((size_t)b * N_OUT + nrowS[row]) * ODIM + col0 + colN] = y;
        }
    }
}

extern "C" void kernel_launch(void* const* d_in, const int* in_sizes, int n_in,
                              void* d_out, int out_size, void* d_ws, size_t ws_size,
                              hipStream_t stream) {
    const float* last    = (const float*)d_in[0];
    const float* W       = (const float*)d_in[1];
    const float* bias    = (const float*)d_in[2];
    const float* alpha   = (const float*)d_in[3];
    const int*   vec     = (const int*)d_in[4];
    const int*   dmap    = (const int*)d_in[5];
    const int*   drev    = (const int*)d_in[6];
    const int*   child_l = (const int*)d_in[7];
    const int*   child_r = (const int*)d_in[8];
    float* out = (float*)d_out;

    // workspace layout
    char* ws = (char*)d_ws;
    int*   counts  = (int*)ws;                                          // 96 ints
    int*   buckets = (int*)(ws + 512);                                  // 16*6*8192 ints = 3 MB
    float* Wt      = (float*)(ws + 512 + (size_t)B_ * NDIR * N_OUT * 4); // 786 KB

    k_zero_counts<<<1, 128, 0, stream>>>(counts);
    k_transpose_w<<<(NDIR * KDIM * ODIM + 255) / 256, 256, 0, stream>>>(W, Wt);
    k_bucketize<<<(B_ * N_OUT + 255) / 256, 256, 0, stream>>>(vec, dmap, counts, buckets);
    k_gemm_tile<<<B_ * NDIR * TILES_MAX, 256, 0, stream>>>(
        last, Wt, bias, alpha, vec, drev, child_l, child_r, counts, buckets, out);
}